// RangeShrink_max_37838661877874
// MI455X (gfx1250) — compile-verified
//
#include <hip/hip_runtime.h>

// RangeShrink_max: out[row] = -max( P[row, hard_label[row,j]]
//                                   for j where hard_label>=0 && j>=sent_idx[row], init 0 )
// Memory-bound (~77 MB @ 23.3 TB/s ~= 3.3 us). One wave32 per row.
// CDNA5 paths used: global_load_async_to_lds_b64 (+ s_wait_asynccnt), global_prefetch.

#define ROWS_PER_BLOCK 8
#define THREADS 256

typedef __attribute__((address_space(3))) int lds_int;

__global__ __launch_bounds__(THREADS)
void topic_masked_max_kernel(const float* __restrict__ P,
                             const int*   __restrict__ hard_label,
                             const int*   __restrict__ sent_idx,
                             float*       __restrict__ out,
                             int B)
{
    __shared__ int slab[ROWS_PER_BLOCK * 64];   // 2 KB of labels for this block

    const int  tid     = threadIdx.x;
    const int  wave    = tid >> 5;               // 8 waves / block
    const int  lane    = tid & 31;
    const long rowBase = (long)blockIdx.x * ROWS_PER_BLOCK;
    const long row     = rowBase + wave;         // wave-uniform

    // ---- async-copy this block's 8x64 int32 labels into LDS (8 B per lane).
    // Wave w's lanes (tid in [32w,32w+32)) copy exactly ints [64w,64w+64),
    // i.e. its own row's labels -> a per-wave s_wait_asynccnt suffices.
    if (row < B) {
        const int* gsrc  = hard_label + rowBase * 64 + (long)tid * 2;
        unsigned   laddr = (unsigned)(unsigned long)((lds_int*)slab + tid * 2);
        asm volatile("global_load_async_to_lds_b64 %0, %1, off"
                     :: "v"(laddr), "v"(gsrc)
                     : "memory");
    }

    if (row >= B) return;                        // wave-uniform exit (tail only)

    // Overlap with the async copy: prefetch the 324 B P row and load sent_idx.
    const float* Prow = P + row * 81;
    __builtin_prefetch(Prow + lane * 3, 0, 0);   // lane*12 B, spans [0, 384) B
    const int s = sent_idx[row];                 // wave-uniform value

    asm volatile("s_wait_asynccnt 0x0" ::: "memory");

    // Lane covers positions j = lane and j = lane + 32.
    const int l0 = slab[wave * 64 + lane];
    const int l1 = slab[wave * 64 + 32 + lane];

    const bool v0 = (l0 >= 0) && (lane >= s);
    const bool v1 = (l1 >= 0) && ((lane + 32) >= s);

    // Unconditional clamped gathers (footprint <= 4 cachelines per row).
    const float p0 = Prow[v0 ? l0 : 0];
    const float p1 = Prow[v1 ? l1 : 0];
    float m = fmaxf(v0 ? p0 : 0.0f, v1 ? p1 : 0.0f);

    // wave32 max reduction (5 xor-shuffle steps).
    #pragma unroll
    for (int off = 16; off > 0; off >>= 1)
        m = fmaxf(m, __shfl_xor(m, off, 32));

    if (lane == 0) out[row] = -m;                // empty set -> -0.0 (== 0.0)
}

extern "C" void kernel_launch(void* const* d_in, const int* in_sizes, int n_in,
                              void* d_out, int out_size, void* d_ws, size_t ws_size,
                              hipStream_t stream)
{
    (void)n_in; (void)out_size; (void)d_ws; (void)ws_size;

    const float* P          = (const float*)d_in[0];   // [B, 81] f32
    const int*   hard_label = (const int*)d_in[1];     // [B, 64] i32
    const int*   sent_idx   = (const int*)d_in[2];     // [B]     i32
    float*       out        = (float*)d_out;           // [B]     f32

    const int B      = in_sizes[2];
    const int blocks = (B + ROWS_PER_BLOCK - 1) / ROWS_PER_BLOCK;

    hipLaunchKernelGGL(topic_masked_max_kernel,
                       dim3(blocks), dim3(THREADS), 0, stream,
                       P, hard_label, sent_idx, out, B);
}